// yolo_loss_57114475102621
// MI455X (gfx1250) — compile-verified
//
#include <hip/hip_runtime.h>
#include <hip/hip_bf16.h>

#define NUM_CLASSES 80
#define CCH 85
#define NA 3
#define BSZ 16
#define MAX_BOXES 32
#define INPUT_SZ 416.0f
#define IGNORE_THRESH 0.5f

// anchors per layer (ANCHOR_MASK applied): layer0 -> [116,90],[156,198],[373,326]
__device__ __constant__ float d_anchors[3][NA][2] = {
    {{116.f, 90.f}, {156.f, 198.f}, {373.f, 326.f}},
    {{30.f, 61.f}, {62.f, 45.f}, {59.f, 119.f}},
    {{10.f, 13.f}, {16.f, 30.f}, {33.f, 23.f}},
};

__device__ __forceinline__ float sigmoidf_(float x) {
    return 1.0f / (1.0f + expf(-x));
}

__device__ __forceinline__ float bce_logits(float x, float t) {
    // softplus(x) - x*t ; softplus(x) = max(x,0) + log1p(exp(-|x|))
    return fmaxf(x, 0.0f) + log1pf(expf(-fabsf(x))) - x * t;
}

// ---------------------------------------------------------------------------
// Kernel 1: gather ground-truth boxes (obj > 0.5) per (layer, batch).
// One block per (layer, batch); order of boxes is irrelevant (max-IoU later).
// ---------------------------------------------------------------------------
__global__ void gather_boxes_kernel(const float* __restrict__ yt0,
                                    const float* __restrict__ yt1,
                                    const float* __restrict__ yt2,
                                    float* __restrict__ boxes,  // [48][32][4]
                                    int* __restrict__ counts) { // [48]
    const int lb = blockIdx.x;          // 0..47
    const int l = lb / BSZ;
    const int b = lb % BSZ;
    const float* yt = (l == 0) ? yt0 : ((l == 1) ? yt1 : yt2);
    const int g = (l == 0) ? 13 : ((l == 1) ? 26 : 52);
    const int ncell = g * g * NA;

    __shared__ float sbox[MAX_BOXES][4];
    __shared__ int scnt;
    if (threadIdx.x == 0) scnt = 0;
    __syncthreads();

    for (int i = threadIdx.x; i < ncell; i += blockDim.x) {
        const float* p = yt + ((size_t)b * ncell + i) * CCH;
        if (p[4] > 0.5f) {
            int k = atomicAdd(&scnt, 1);
            if (k < MAX_BOXES) {
                sbox[k][0] = p[0];
                sbox[k][1] = p[1];
                sbox[k][2] = p[2];
                sbox[k][3] = p[3];
            }
        }
    }
    __syncthreads();
    int cnt = scnt;
    if (cnt > MAX_BOXES) cnt = MAX_BOXES;
    if (threadIdx.x == 0) counts[lb] = cnt;
    for (int k = threadIdx.x; k < MAX_BOXES * 4; k += blockDim.x) {
        boxes[(size_t)lb * MAX_BOXES * 4 + k] = (k < cnt * 4) ? sbox[k >> 2][k & 3] : 0.0f;
    }
}

// ---------------------------------------------------------------------------
// Kernel 2: per-layer loss. Each block writes 4 partial sums (deterministic
// in-block tree reduction). Lanes map to consecutive (y,x) -> coalesced feats.
// ---------------------------------------------------------------------------
__global__ void yolo_layer_loss_kernel(const float* __restrict__ feats,
                                       const float* __restrict__ ytrue,
                                       const float* __restrict__ boxes,  // [BSZ][32][4] this layer
                                       const int* __restrict__ counts,   // [BSZ] this layer
                                       float* __restrict__ partials,     // [gridDim.x][4]
                                       int g, int layer) {
    const int ghw = g * g;
    const int ncell_total = BSZ * NA * ghw;
    const float fg = (float)g;

    __shared__ float sbox[BSZ][MAX_BOXES][4]; // 8 KB
    __shared__ int scnt[BSZ];
    for (int k = threadIdx.x; k < BSZ * MAX_BOXES * 4; k += blockDim.x)
        ((float*)sbox)[k] = boxes[k];
    for (int k = threadIdx.x; k < BSZ; k += blockDim.x) scnt[k] = counts[k];
    __syncthreads();

    float axy = 0.f, awh = 0.f, aconf = 0.f, acls = 0.f;

    for (int i = blockIdx.x * blockDim.x + threadIdx.x; i < ncell_total;
         i += gridDim.x * blockDim.x) {
        const int yx = i % ghw;
        const int ba = i / ghw;          // b*NA + a
        const int a = ba % NA;
        const int b = ba / NA;
        const int gx = yx % g;
        const int gy = yx / g;

        const float* fp = feats + (size_t)ba * CCH * ghw + yx;  // channel stride = ghw
        const float* tp = ytrue + ((size_t)b * ghw * NA + (size_t)yx * NA + a) * CCH;

        const float r0 = fp[0 * ghw], r1 = fp[1 * ghw], r2 = fp[2 * ghw],
                    r3 = fp[3 * ghw], r4 = fp[4 * ghw];
        const float t0 = tp[0], t1 = tp[1], t2 = tp[2], t3 = tp[3], obj = tp[4];

        const float aw = d_anchors[layer][a][0];
        const float ah = d_anchors[layer][a][1];

        // predicted box (normalized)
        const float px = (sigmoidf_(r0) + (float)gx) / fg;
        const float py = (sigmoidf_(r1) + (float)gy) / fg;
        const float pw = expf(r2) * aw / INPUT_SZ;
        const float ph = expf(r3) * ah / INPUT_SZ;

        // best IoU against the batch's true boxes
        float best = 0.0f;
        const int cnt = scnt[b];
        const float pxmin = px - pw * 0.5f, pxmax = px + pw * 0.5f;
        const float pymin = py - ph * 0.5f, pymax = py + ph * 0.5f;
        const float parea = pw * ph;
        for (int k = 0; k < cnt; ++k) {
            const float bx = sbox[b][k][0], by = sbox[b][k][1];
            const float bw = sbox[b][k][2], bh = sbox[b][k][3];
            float ix = fminf(pxmax, bx + bw * 0.5f) - fmaxf(pxmin, bx - bw * 0.5f);
            float iy = fminf(pymax, by + bh * 0.5f) - fmaxf(pymin, by - bh * 0.5f);
            ix = fmaxf(ix, 0.0f);
            iy = fmaxf(iy, 0.0f);
            const float inter = ix * iy;
            const float uni = parea + bw * bh - inter;
            best = fmaxf(best, inter / uni);
        }
        const float ignore = (best < IGNORE_THRESH) ? 1.0f : 0.0f;

        // conf loss for every cell
        const float cb = bce_logits(r4, obj);
        aconf += cb * obj + (1.0f - obj) * cb * ignore;

        if (obj > 0.0f) {
            const float txy0 = t0 * fg - (float)gx;
            const float txy1 = t1 * fg - (float)gy;
            const float twh0 = logf(t2 / aw * INPUT_SZ);
            const float twh1 = logf(t3 / ah * INPUT_SZ);
            const float scale = 2.0f - t2 * t3;
            axy += obj * scale * (bce_logits(r0, txy0) + bce_logits(r1, txy1));
            const float dw = r2 - twh0, dh = r3 - twh1;
            awh += obj * scale * (dw * dw + dh * dh);
            float s = 0.0f;
            for (int c = 5; c < CCH; ++c) s += bce_logits(fp[(size_t)c * ghw], tp[c]);
            acls += obj * s;
        }
    }

    // deterministic block tree reduction
    __shared__ float red[4][256];
    const int tid = threadIdx.x;
    red[0][tid] = axy;
    red[1][tid] = awh;
    red[2][tid] = aconf;
    red[3][tid] = acls;
    __syncthreads();
    for (int s = 128; s > 0; s >>= 1) {
        if (tid < s) {
            red[0][tid] += red[0][tid + s];
            red[1][tid] += red[1][tid + s];
            red[2][tid] += red[2][tid + s];
            red[3][tid] += red[3][tid + s];
        }
        __syncthreads();
    }
    if (tid < 4) partials[(size_t)blockIdx.x * 4 + tid] = red[tid][0];
}

// ---------------------------------------------------------------------------
// Kernel 3: final reduction via V_WMMA_F32_16X16X4_F32.
// D = ones(16x4) x B(4x16) accumulated over row-chunks -> column sums.
// Column-sum is invariant to the exact K-row->lane assignment; only the
// N = lane%16 column mapping matters (documented C/D/B layout).
// One wave, no divergence around the WMMA (EXEC all-1s).
// ---------------------------------------------------------------------------
typedef float v2f __attribute__((ext_vector_type(2)));
typedef float v8f __attribute__((ext_vector_type(8)));

__global__ void finalize_wmma_kernel(const float* __restrict__ partials,
                                     int nrows,  // multiple of 4
                                     float* __restrict__ out) {
    const int lane = threadIdx.x;       // 0..31
    const int n = lane & 15;            // column
    const int hi = lane >> 4;           // lane half
    const float mask = (n < 4) ? 1.0f : 0.0f;

    v2f a;
    a[0] = 1.0f;
    a[1] = 1.0f;
    v8f c = {};

    for (int t = 0; t < nrows; t += 4) {
        const int r0 = t + hi;
        const int r1 = t + 2 + hi;
        v2f bv;
        bv[0] = partials[r0 * 4 + (n & 3)] * mask;
        bv[1] = partials[r1 * 4 + (n & 3)] * mask;
        c = __builtin_amdgcn_wmma_f32_16x16x4_f32(
            /*neg_a=*/false, a, /*neg_b=*/false, bv,
            /*c_mod=*/(short)0, c, /*reuse_a=*/false, /*reuse_b=*/false);
    }

    __shared__ float s[4];
    if (lane < 4) s[lane] = c[0];  // D[0][n], n = lane
    __syncthreads();
    if (lane == 0) {
        const float inv_bs = 1.0f / (float)BSZ;
        const float xy = s[0] * inv_bs;
        const float wh = s[1] * inv_bs;
        const float cf = s[2] * inv_bs;
        const float cl = s[3] * inv_bs;
        out[0] = xy + wh + cf + cl;
        out[1] = xy;
        out[2] = wh;
        out[3] = cf;
        out[4] = cl;
    }
}

// ---------------------------------------------------------------------------
extern "C" void kernel_launch(void* const* d_in, const int* in_sizes, int n_in,
                              void* d_out, int out_size, void* d_ws, size_t ws_size,
                              hipStream_t stream) {
    // setup_inputs() dict order: feats0, ytrue0, feats1, ytrue1, feats2, ytrue2
    const float* f0 = (const float*)d_in[0];
    const float* y0 = (const float*)d_in[1];
    const float* f1 = (const float*)d_in[2];
    const float* y1 = (const float*)d_in[3];
    const float* f2 = (const float*)d_in[4];
    const float* y2 = (const float*)d_in[5];
    float* out = (float*)d_out;

    char* ws = (char*)d_ws;
    float* boxes = (float*)ws;                                     // 48*32*4 floats = 24576 B
    int* counts = (int*)(ws + 48 * MAX_BOXES * 4 * sizeof(float)); // 48 ints
    float* partials = (float*)(ws + 48 * MAX_BOXES * 4 * sizeof(float) + 256); // 336*4 floats

    // 1) gather true boxes per (layer, batch)
    gather_boxes_kernel<<<48, 256, 0, stream>>>(y0, y1, y2, boxes, counts);

    // 2) per-layer loss partials (rows: L0=16, L1=64, L2=256 -> 336 total)
    yolo_layer_loss_kernel<<<16, 256, 0, stream>>>(
        f0, y0, boxes + 0 * BSZ * MAX_BOXES * 4, counts + 0 * BSZ,
        partials + 0 * 4, 13, 0);
    yolo_layer_loss_kernel<<<64, 256, 0, stream>>>(
        f1, y1, boxes + 1 * BSZ * MAX_BOXES * 4, counts + 1 * BSZ,
        partials + 16 * 4, 26, 1);
    yolo_layer_loss_kernel<<<256, 256, 0, stream>>>(
        f2, y2, boxes + 2 * BSZ * MAX_BOXES * 4, counts + 2 * BSZ,
        partials + 80 * 4, 52, 2);

    // 3) WMMA-based deterministic final reduction (336 rows, multiple of 4)
    finalize_wmma_kernel<<<1, 32, 0, stream>>>(partials, 336, out);
}